// RSR_65317862637910
// MI455X (gfx1250) — compile-verified
//
#include <hip/hip_runtime.h>
#include <math.h>

// Graph-attention edge softmax, factored form for MI455X (gfx1250, wave32).
//
//  score_e = p_src[src] + p_dst[dst] + p_et[etype] + b   (separable dot product)
//  aggr[n] = emb[n] * segment_sum(weight)_n              (dst_emb constant per segment)
//
// Pipeline:
//  1) init scratch
//  2) WMMA f32 16x16x4 GEMM: [p_src|p_dst] = emb @ [w0|w1|0...]   (v_wmma path)
//  3) p_et = edge_embeddings @ w2  (64 rows, trivial)
//  4) edge pass 1: degrees + segment-max(score) via monotonic-uint atomicMax
//  5) edge pass 2: denom[dst] += exp(s - smax[dst])
//  6) edge pass 3: wsum[dst]  += exp(s - smax)/denom/clip(outdeg,1)
//  7) out[n,:] = indeg[n]>0 ? emb[n,:]*wsum[n] : emb[n,:]

typedef __attribute__((ext_vector_type(2))) float v2f;
typedef __attribute__((ext_vector_type(8))) float v8f;

#define D_FEAT 128

// ---- monotonic uint encoding of float for atomicMax-based segment max ----
__device__ __forceinline__ unsigned f2key(float f) {
    unsigned b = __float_as_uint(f);
    return (b & 0x80000000u) ? ~b : (b | 0x80000000u);
}
__device__ __forceinline__ float key2f(unsigned k) {
    return __uint_as_float((k & 0x80000000u) ? (k & 0x7fffffffu) : ~k);
}

__global__ void gat_init(unsigned* __restrict__ outdeg, unsigned* __restrict__ indeg,
                         unsigned* __restrict__ smaxk, float* __restrict__ denom,
                         float* __restrict__ wsum, int N) {
    int i = blockIdx.x * blockDim.x + threadIdx.x;
    if (i < N) {
        outdeg[i] = 0u; indeg[i] = 0u; smaxk[i] = 0u;  // key 0 == "never updated"
        denom[i] = 0.0f; wsum[i] = 0.0f;
    }
}

// P = emb(Nx128) @ B(128x16), B col0 = w[0:128], col1 = w[128:256], cols 2..15 = 0.
// One wave per 16-row tile; 32 K-steps of V_WMMA_F32_16X16X4_F32.
// A layout (16x4 f32): lane m=lane&15 holds row M=m; half=lane>>4 selects K pair
//   {VGPR0,VGPR1} = {K=2*half, K=2*half+1} within each 4-wide K step.
// B layout (4x16 f32): lane n=lane&15 is column; same half/K striping as A.
// C/D layout: VGPR v holds M = v + 8*half, N = lane&15.
//
// B load is branch-free: every lane loads a valid 8B chunk of w (lanes >= 2
// read the w[256:384] region) and multiplies by a 0/1 mask, so the inner loop
// is load_b64 + load_b64 + wmma with no exec-mask manipulation.
__global__ void gat_proj_wmma(const float* __restrict__ emb, const float* __restrict__ w,
                              float* __restrict__ psrc, float* __restrict__ pdst, int N) {
    int lane = threadIdx.x & 31;
    int wave = threadIdx.x >> 5;
    int tile = blockIdx.x * (blockDim.x >> 5) + wave;
    int ntiles = (N + 15) >> 4;
    if (tile >= ntiles) return;           // wave-uniform exit: EXEC all-ones for WMMA

    int half = lane >> 4;                 // 0 or 1
    int m    = lane & 15;                 // A row / B-D column
    int row  = tile * 16 + m;
    int rcl  = row < N ? row : N - 1;     // clamp tail reads
    const float* arow = emb + (size_t)rcl * D_FEAT;
    float bmask = (m < 2) ? 1.0f : 0.0f;
    const float* bcol = w + (size_t)(m < 2 ? m : 2) * D_FEAT; // always in-bounds

    v8f c = {0.f, 0.f, 0.f, 0.f, 0.f, 0.f, 0.f, 0.f};
#pragma unroll 8
    for (int kk = 0; kk < D_FEAT / 4; ++kk) {
        int kb = kk * 4 + half * 2;       // even -> 8B-aligned v2f loads
        v2f a  = *(const v2f*)(arow + kb);
        v2f bl = *(const v2f*)(bcol + kb);
        v2f bv; bv.x = bl.x * bmask; bv.y = bl.y * bmask;
        c = __builtin_amdgcn_wmma_f32_16x16x4_f32(
                /*neg_a=*/false, a, /*neg_b=*/false, bv,
                /*c_mod=*/(short)0, c, /*reuse_a=*/false, /*reuse_b=*/false);
    }

#pragma unroll
    for (int v = 0; v < 8; ++v) {
        int orow = tile * 16 + v + half * 8;
        if (orow < N) {
            if (m == 0)      psrc[orow] = c[v];
            else if (m == 1) pdst[orow] = c[v];
        }
    }
}

__global__ void gat_proj_et(const float* __restrict__ ee, const float* __restrict__ w,
                            float* __restrict__ pet, int R) {
    int r = blockIdx.x * blockDim.x + threadIdx.x;
    if (r >= R) return;
    const float* er = ee + (size_t)r * D_FEAT;
    const float* we = w + 2 * D_FEAT;
    float s = 0.0f;
#pragma unroll 8
    for (int k = 0; k < D_FEAT; ++k) s += er[k] * we[k];
    pet[r] = s;
}

__device__ __forceinline__ float edge_score(int e, const int* src, const int* dst,
                                            const int* et, const float* psrc,
                                            const float* pdst, const float* pet,
                                            float bias, int& d_i) {
    int s_i = src[e]; d_i = dst[e]; int t_i = et[e];
    float s = psrc[s_i] + pdst[d_i] + pet[t_i] + bias;
    return s > 0.0f ? s : 0.2f * s;       // leaky_relu(0.2)
}

__global__ void gat_edge_max(const int* __restrict__ src, const int* __restrict__ dst,
                             const int* __restrict__ et, const float* __restrict__ psrc,
                             const float* __restrict__ pdst, const float* __restrict__ pet,
                             const float* __restrict__ bptr, unsigned* __restrict__ outdeg,
                             unsigned* __restrict__ indeg, unsigned* __restrict__ smaxk, int E) {
    int e = blockIdx.x * blockDim.x + threadIdx.x;
    if (e >= E) return;
    atomicAdd(&outdeg[src[e]], 1u);
    int d_i;
    float s = edge_score(e, src, dst, et, psrc, pdst, pet, bptr[0], d_i);
    atomicAdd(&indeg[d_i], 1u);
    atomicMax(&smaxk[d_i], f2key(s));
}

__global__ void gat_edge_denom(const int* __restrict__ src, const int* __restrict__ dst,
                               const int* __restrict__ et, const float* __restrict__ psrc,
                               const float* __restrict__ pdst, const float* __restrict__ pet,
                               const float* __restrict__ bptr, const unsigned* __restrict__ smaxk,
                               float* __restrict__ denom, int E) {
    int e = blockIdx.x * blockDim.x + threadIdx.x;
    if (e >= E) return;
    int d_i;
    float s = edge_score(e, src, dst, et, psrc, pdst, pet, bptr[0], d_i);
    float ex = expf(s - key2f(smaxk[d_i]));
    atomicAdd(&denom[d_i], ex);
}

__global__ void gat_edge_wsum(const int* __restrict__ src, const int* __restrict__ dst,
                              const int* __restrict__ et, const float* __restrict__ psrc,
                              const float* __restrict__ pdst, const float* __restrict__ pet,
                              const float* __restrict__ bptr, const unsigned* __restrict__ smaxk,
                              const float* __restrict__ denom, const unsigned* __restrict__ outdeg,
                              float* __restrict__ wsum, int E) {
    int e = blockIdx.x * blockDim.x + threadIdx.x;
    if (e >= E) return;
    int d_i;
    float s = edge_score(e, src, dst, et, psrc, pdst, pet, bptr[0], d_i);
    float ex = expf(s - key2f(smaxk[d_i]));
    unsigned odr = outdeg[d_i];
    float od = (float)(odr > 1u ? odr : 1u);   // jnp.clip(out_deg, 1.0)
    atomicAdd(&wsum[d_i], ex / (denom[d_i] * od));
}

__global__ void gat_final(const float* __restrict__ emb, const unsigned* __restrict__ indeg,
                          const float* __restrict__ wsum, float* __restrict__ out, int N) {
    int i = blockIdx.x * blockDim.x + threadIdx.x;   // over N * D/4 float4 elements
    int total = N * (D_FEAT / 4);
    if (i >= total) return;
    int node = i >> 5;                               // D/4 == 32 float4 per node
    float4 v = ((const float4*)emb)[i];
    float sc = (indeg[node] > 0u) ? wsum[node] : 1.0f;
    v.x *= sc; v.y *= sc; v.z *= sc; v.w *= sc;
    ((float4*)out)[i] = v;
}

extern "C" void kernel_launch(void* const* d_in, const int* in_sizes, int n_in,
                              void* d_out, int out_size, void* d_ws, size_t ws_size,
                              hipStream_t stream) {
    const float* emb   = (const float*)d_in[0];
    const float* eemb  = (const float*)d_in[1];
    const float* w     = (const float*)d_in[2];
    const float* bptr  = (const float*)d_in[3];
    const int*   src   = (const int*)d_in[4];
    const int*   dst   = (const int*)d_in[5];
    const int*   etype = (const int*)d_in[6];

    const int D = D_FEAT;
    const int N = in_sizes[0] / D;
    const int R = in_sizes[1] / D;
    const int E = in_sizes[4];

    // Scratch layout: 8 N-sized slots (needs 8*N*4 bytes; 320 KB for N=10000).
    float* ws = (float*)d_ws;
    unsigned* outdeg = (unsigned*)(ws + 0 * (size_t)N);
    unsigned* indeg  = (unsigned*)(ws + 1 * (size_t)N);
    float*    psrc   =             ws + 2 * (size_t)N;
    float*    pdst   =             ws + 3 * (size_t)N;
    float*    pet    =             ws + 4 * (size_t)N;   // uses first R entries
    unsigned* smaxk  = (unsigned*)(ws + 5 * (size_t)N);
    float*    denom  =             ws + 6 * (size_t)N;
    float*    wsum   =             ws + 7 * (size_t)N;

    gat_init<<<(N + 255) / 256, 256, 0, stream>>>(outdeg, indeg, smaxk, denom, wsum, N);

    int ntiles = (N + 15) / 16;
    gat_proj_wmma<<<(ntiles + 7) / 8, 256, 0, stream>>>(emb, w, psrc, pdst, N);
    gat_proj_et<<<(R + 63) / 64, 64, 0, stream>>>(eemb, w, pet, R);

    int eblocks = (E + 255) / 256;
    gat_edge_max<<<eblocks, 256, 0, stream>>>(src, dst, etype, psrc, pdst, pet, bptr,
                                              outdeg, indeg, smaxk, E);
    gat_edge_denom<<<eblocks, 256, 0, stream>>>(src, dst, etype, psrc, pdst, pet, bptr,
                                                smaxk, denom, E);
    gat_edge_wsum<<<eblocks, 256, 0, stream>>>(src, dst, etype, psrc, pdst, pet, bptr,
                                               smaxk, denom, outdeg, wsum, E);

    gat_final<<<(N * (D / 4) + 255) / 256, 256, 0, stream>>>(emb, indeg, wsum,
                                                             (float*)d_out, N);
}